// Quantize_bi_11905649344702
// MI455X (gfx1250) — compile-verified
//
#include <hip/hip_runtime.h>
#include <hip/hip_bf16.h>
#include <float.h>

typedef float v2f __attribute__((ext_vector_type(2)));
typedef float v4f __attribute__((ext_vector_type(4)));
typedef float v8f __attribute__((ext_vector_type(8)));

#define DIM      64
#define NEMB     1024
#define POSD     16
#define POSE     128
#define NROWS    65536          // 64*32*32
#define RPB      128            // rows per block
#define THREADS  256            // 8 wave32
#define NTILES   64             // 1024/16 column tiles
#define NPAIRS   8              // 16 k-steps of 4 -> 8 pairs

// ---- workspace layout (float offsets) ----
#define WS_CNORM 0              // 1024
#define WS_PART  1024           // 512 block partials
#define WS_BPACK 2048           // 64 tiles * 8 pairs * 32 lanes * 4 = 65536
#define WS_COLT  67584          // cols^T [j][d] = 65536
// total ws: 133120 floats = 532 KB

// ---- output layout (float offsets, tuple concatenated flat) ----
#define OUT_Q    0              // 4194304 quantize_st
#define OUT_DIFF 4194304        // 64
#define OUT_IND  4194368        // 65536 (argmin, stored as float)
#define OUT_EMB  4259904        // 65536 embed_m

// Two chained 16x16x4 f32 WMMAs consuming one float4 B-fragment (2 k-steps).
// P must be a compile-time constant after unrolling -> static afrag indexing.
#define MM2(P, B, ACC) do {                                                   \
    v2f _b0; _b0.x = (B).x; _b0.y = (B).y;                                    \
    v2f _b1; _b1.x = (B).z; _b1.y = (B).w;                                    \
    (ACC) = __builtin_amdgcn_wmma_f32_16x16x4_f32(false, afrag[2*(P)],        \
                false, _b0, (short)0, (ACC), false, false);                   \
    (ACC) = __builtin_amdgcn_wmma_f32_16x16x4_f32(false, afrag[2*(P)+1],      \
                false, _b1, (short)0, (ACC), false, false);                   \
  } while (0)

// Kernel 1: mask codebook, emit embed_m + cnorm + transposed cols +
// WMMA-B fragments pre-swizzled into per-lane layout.
// B 4x16 f32 layout mirrors A 16x4: lanes 0-15 hold K=4s+{0,1},
// lanes 16-31 hold K=4s+{2,3} (VGPR0/VGPR1).
__global__ __launch_bounds__(256)
void vq_prep(const float* __restrict__ embed, float* __restrict__ out,
             float* __restrict__ ws) {
  int j = blockIdx.x * blockDim.x + threadIdx.x;   // code column 0..1023
  if (j >= NEMB) return;
  float col[DIM];
  float cn = 0.f;
  #pragma unroll
  for (int d = 0; d < DIM; ++d) {
    float e = embed[d * NEMB + j];
    bool keep = (d < DIM - POSD) ? (j < NEMB - POSE) : (j >= NEMB - POSE);
    float m = keep ? e : 0.f;
    col[d] = m;
    cn += m * m;
    out[OUT_EMB + d * NEMB + j] = m;   // embed_m output
    ws[WS_COLT + j * DIM + d] = m;     // transposed for coalesced gather
  }
  ws[WS_CNORM + j] = cn;

  int tile = j >> 4, N = j & 15;
  v4f* bp = (v4f*)(ws + WS_BPACK);
  #pragma unroll
  for (int p = 0; p < NPAIRS; ++p) {
    int k0 = 8 * p;                        // pair p covers dims 8p..8p+7
    int base = (tile * NPAIRS + p) * 32;
    v4f lo = { col[k0 + 0], col[k0 + 1], col[k0 + 4], col[k0 + 5] };  // lanes 0-15
    v4f hi = { col[k0 + 2], col[k0 + 3], col[k0 + 6], col[k0 + 7] };  // lanes 16-31
    bp[base + N]      = lo;
    bp[base + N + 16] = hi;
  }
}

// Kernel 2: WMMA distance GEMM + argmin + quantize gather + diff partials.
__global__ __launch_bounds__(THREADS)
void vq_main(const float* __restrict__ inp, float* __restrict__ out,
             float* __restrict__ ws) {
  __shared__ float lds_in[RPB * DIM];   // 32 KB input tile
  __shared__ float lds_cn[NEMB];        // 4 KB code norms
  __shared__ int   lds_idx[RPB];
  __shared__ float lds_red[THREADS];

  const int tid = threadIdx.x;
  const int blk = blockIdx.x;

  // stage 128x64 f32 tile (coalesced b128) + cnorm into LDS
  {
    const v4f* src = (const v4f*)(inp + (size_t)blk * RPB * DIM);
    v4f* dst = (v4f*)lds_in;
    #pragma unroll
    for (int i = 0; i < (RPB * DIM / 4) / THREADS; ++i)
      dst[tid + i * THREADS] = src[tid + i * THREADS];
    #pragma unroll
    for (int i = 0; i < NEMB / THREADS; ++i)
      lds_cn[tid + i * THREADS] = ws[WS_CNORM + tid + i * THREADS];
  }
  __syncthreads();

  const int wave   = tid >> 5;          // wave32
  const int lane   = tid & 31;
  const int lane15 = lane & 15;
  const int khalf  = lane >> 4;         // 0: K+{0,1}, 1: K+{2,3}

  // A fragments: row = wave*16 + lane15, step s -> dims 4s + 2*khalf + {0,1}
  v2f afrag[16];
  {
    const float* rowp = lds_in + (wave * 16 + lane15) * DIM;
    #pragma unroll
    for (int s = 0; s < 16; ++s)
      afrag[s] = *(const v2f*)(rowp + 4 * s + 2 * khalf);
  }

  const v4f* bp = (const v4f*)(ws + WS_BPACK);

  float best[8]; int bidx[8];
  #pragma unroll
  for (int r = 0; r < 8; ++r) { best[r] = FLT_MAX; bidx[r] = 0; }

  auto scoreUpdate = [&](int tile, const v8f& acc) {
    const float cn = lds_cn[tile * 16 + lane15];
    const int   j  = tile * 16 + lane15;
    #pragma unroll
    for (int r = 0; r < 8; ++r) {
      float score = fmaf(-2.f, acc[r], cn);   // ||c||^2 - 2 x.c  (||x||^2 constant)
      if (score < best[r]) { best[r] = score; bidx[r] = j; }
    }
  };
  auto loadB6 = [&](v4f b[6], int tile) {
    #pragma unroll
    for (int p = 0; p < 6; ++p)
      b[p] = bp[(tile * NPAIRS + p) * 32 + lane];   // coalesced b128
  };

  // ---- tiles 0..55: dims 0..47 (block-diagonal mask), double-buffered B ----
  {
    v4f bA[6], bB[6];
    loadB6(bA, 0);
    for (int t2 = 0; t2 < 28; ++t2) {
      const int tile = 2 * t2;
      loadB6(bB, tile + 1);                 // prefetch next while computing cur
      {
        v8f acc = {};
        #pragma unroll
        for (int p = 0; p < 6; ++p) MM2(p, bA[p], acc);
        scoreUpdate(tile, acc);
      }
      if (t2 < 27) loadB6(bA, tile + 2);    // prefetch tile+2 while computing
      {
        v8f acc = {};
        #pragma unroll
        for (int p = 0; p < 6; ++p) MM2(p, bB[p], acc);
        scoreUpdate(tile + 1, acc);
      }
    }
  }

  // ---- tiles 56..63: dims 48..63 (pairs 6,7 -> afrag[12..15]) ----
  #pragma unroll 2
  for (int tile = 56; tile < 64; ++tile) {
    v4f b6 = bp[(tile * NPAIRS + 6) * 32 + lane];
    v4f b7 = bp[(tile * NPAIRS + 7) * 32 + lane];
    v8f acc = {};
    MM2(6, b6, acc);
    MM2(7, b7, acc);
    scoreUpdate(tile, acc);
  }

  // argmin across the 16-lane halves (D layout: N = lane&15, M = r + 8*khalf)
  #pragma unroll
  for (int r = 0; r < 8; ++r) {
    float v = best[r]; int id = bidx[r];
    #pragma unroll
    for (int m = 1; m < 16; m <<= 1) {
      float ov = __shfl_xor(v, m, 32);
      int   oid = __shfl_xor(id, m, 32);
      if (ov < v || (ov == v && oid < id)) { v = ov; id = oid; }
    }
    best[r] = v; bidx[r] = id;
  }

  if (lane15 == 0) {
    const int rowOff = wave * 16 + khalf * 8;
    #pragma unroll
    for (int r = 0; r < 8; ++r) {
      const int row = rowOff + r;
      lds_idx[row] = bidx[r];
      out[OUT_IND + (size_t)blk * RPB + row] = (float)bidx[r];
    }
  }
  __syncthreads();

  // gather quantize (straight-through) + diff partial, input reused from LDS
  const float* colT = ws + WS_COLT;
  float local = 0.f;
  #pragma unroll
  for (int i = 0; i < (RPB * DIM) / THREADS; ++i) {
    const int e = tid + i * THREADS;
    const int row = e >> 6, d = e & 63;
    const int j = lds_idx[row];
    const float q = colT[j * DIM + d];   // coalesced within a row
    const float x = lds_in[e];
    const float df = q - x;              // diff uses (quantize - input)
    out[OUT_Q + (size_t)blk * RPB * DIM + e] = x + df;  // x + (q - x), matches ref fp
    local += df * df;
  }

  // deterministic fixed-order tree reduction
  lds_red[tid] = local;
  __syncthreads();
  for (int s = THREADS / 2; s > 0; s >>= 1) {
    if (tid < s) lds_red[tid] += lds_red[tid + s];
    __syncthreads();
  }
  if (tid == 0) ws[WS_PART + blk] = lds_red[0];
}

// Kernel 3: per-image mean from block partials (8 blocks per image, fixed order).
__global__ void vq_diff(const float* __restrict__ ws, float* __restrict__ out) {
  int b = threadIdx.x;
  if (b < 64) {
    float s = 0.f;
    #pragma unroll
    for (int k = 0; k < 8; ++k) s += ws[WS_PART + b * 8 + k];
    out[OUT_DIFF + b] = s * (1.f / 65536.f);  // mean over 32*32*64
  }
}

extern "C" void kernel_launch(void* const* d_in, const int* in_sizes, int n_in,
                              void* d_out, int out_size, void* d_ws, size_t ws_size,
                              hipStream_t stream) {
  (void)in_sizes; (void)n_in; (void)out_size; (void)ws_size;
  const float* inp   = (const float*)d_in[0];   // (64,32,32,64) f32
  const float* embed = (const float*)d_in[1];   // (64,1024) f32
  // d_in[2] = bi, fixed to 1 by the harness (output shapes depend on it)
  float* out = (float*)d_out;
  float* ws  = (float*)d_ws;

  vq_prep<<<4, 256, 0, stream>>>(embed, out, ws);
  vq_main<<<NROWS / RPB, THREADS, 0, stream>>>(inp, out, ws);
  vq_diff<<<1, 64, 0, stream>>>(ws, out);
}